// GCN_layer_54554674594289
// MI455X (gfx1250) — compile-verified
//
#include <hip/hip_runtime.h>

typedef float v2f __attribute__((ext_vector_type(2)));
typedef float v8f __attribute__((ext_vector_type(8)));

#define D 128            // feature dim (D_IN == D_OUT == 128)
#define WP 132           // LDS pitch for W rows: 132 % 64 == 4 -> conflict-free
                         // b64 fragment reads AND conflict-free staging stores

// ---------------------------------------------------------------------------
// Kernel 1: zero the aggregation buffer (d_out doubles as agg scratch).
// ---------------------------------------------------------------------------
__global__ void gcn_zero_kernel(float4* __restrict__ p, int n4) {
    int i = blockIdx.x * blockDim.x + threadIdx.x;
    int stride = gridDim.x * blockDim.x;
    float4 z = make_float4(0.f, 0.f, 0.f, 0.f);
    for (; i < n4; i += stride) p[i] = z;
}

// ---------------------------------------------------------------------------
// Kernel 2: edge scatter  agg[r] += v * X[c].
// One wave32 per edge: lane l owns feature elements [4l, 4l+4).
// X row gather is a coalesced 512B read (X = 51.2MB, resident in 192MB L2).
// Scatter lowers to native global_atomic_add_f32 (relaxed, device scope,
// no return -> STOREcnt path, no VGPR writeback).
// ---------------------------------------------------------------------------
__global__ void __launch_bounds__(256)
gcn_scatter_kernel(const float* __restrict__ X,
                   const int*   __restrict__ edge_row,
                   const int*   __restrict__ edge_col,
                   const float* __restrict__ edge_val,
                   float*       __restrict__ agg,
                   int E) {
    int wid  = (blockIdx.x * blockDim.x + threadIdx.x) >> 5;
    int lane = threadIdx.x & 31;
    if (wid >= E) return;

    int   r = edge_row[wid];
    int   c = edge_col[wid];
    float v = edge_val[wid];

    const float4* xrow = (const float4*)(X + (size_t)c * D);
    float4 x = xrow[lane];

    float* dst = agg + (size_t)r * D + lane * 4;
    __hip_atomic_fetch_add(dst + 0, v * x.x, __ATOMIC_RELAXED, __HIP_MEMORY_SCOPE_AGENT);
    __hip_atomic_fetch_add(dst + 1, v * x.y, __ATOMIC_RELAXED, __HIP_MEMORY_SCOPE_AGENT);
    __hip_atomic_fetch_add(dst + 2, v * x.z, __ATOMIC_RELAXED, __HIP_MEMORY_SCOPE_AGENT);
    __hip_atomic_fetch_add(dst + 3, v * x.w, __ATOMIC_RELAXED, __HIP_MEMORY_SCOPE_AGENT);
}

// ---------------------------------------------------------------------------
// Kernel 3: dense GEMM  out = agg @ W^T + b  via V_WMMA_F32_16X16X4_F32.
// In-place on d_out: each wave reads only its own 16 agg rows (all reads
// precede its stores in program order), so no cross-wave hazard.
//
// Block = 256 threads (8 waves); wave w owns rows [blk*128 + 16w, +16) and
// all 128 output columns (8 col-tiles of 16). K = 128 in 32 steps of 4.
//
// A fragments: unconditional global_load_b64 from a row pointer clamped to
// a valid row — garbage in out-of-range A rows only corrupts D rows that
// are never stored, so no EXEC masking in the hot loop.
//
// B fragments: W kept ROW-MAJOR in LDS with pitch WP=132 (== 4 mod 64).
// b = Wlds[o*WP + k0 .. k0+1] is one aligned ds_load_b64 into an even VGPR
// pair; across the wave the 64 accessed dwords land on 64 distinct banks.
// ---------------------------------------------------------------------------
__global__ void __launch_bounds__(256)
gcn_gemm_wmma_kernel(const float* __restrict__ Wmat,   // [D_OUT, D_IN] row-major
                     const float* __restrict__ bias,   // [D_OUT]
                     float*       __restrict__ out,    // in: agg, out: result
                     int nrows) {
    __shared__ float Wlds[D * WP];   // padded row-major copy of W, ~66 KB

    const int tid = threadIdx.x;

    // Stage W: coalesced float4 global reads, ds_store_b128 to consecutive
    // dwords (conflict-free). 4096 float4s, 16 iterations per thread.
    {
        const float4* Wv = (const float4*)Wmat;
        for (int i4 = tid; i4 < (D * D) / 4; i4 += 256) {
            int o = i4 >> 5;          // row of W (output channel)
            int k = (i4 & 31) * 4;    // col of W (input channel)
            *(float4*)&Wlds[o * WP + k] = Wv[i4];
        }
    }
    __syncthreads();

    const int wave = tid >> 5;
    const int lane = tid & 31;
    const int lr   = lane & 15;   // row-within-tile / col-within-tile
    const int hi   = lane >> 4;   // half-wave select

    const int row_base = blockIdx.x * 128 + wave * 16;
    const int row      = row_base + lr;
    // Clamp to a valid row: loads are always in-bounds, never masked.
    const int rowc     = row < nrows ? row : (nrows - 1);
    const float* arow  = out + (size_t)rowc * D;

    v8f acc[8];
#pragma unroll
    for (int t = 0; t < 8; ++t)
        acc[t] = (v8f){0.f, 0.f, 0.f, 0.f, 0.f, 0.f, 0.f, 0.f};

#pragma unroll 8
    for (int kk = 0; kk < D / 4; ++kk) {
        // A fragment (16x4 f32): v0 = K(k0), v1 = K(k0+1); upper half-wave
        // holds K+2/K+3 of the same 16 rows. Single global_load_b64.
        const int k0 = kk * 4 + hi * 2;
        v2f a = *(const v2f*)(arow + k0);

#pragma unroll
        for (int t = 0; t < 8; ++t) {
            // B fragment (4x16 f32): B[k][o] = W[o][k]; one ds_load_b64.
            const int o = t * 16 + lr;
            v2f b = *(const v2f*)&Wlds[o * WP + k0];
            acc[t] = __builtin_amdgcn_wmma_f32_16x16x4_f32(
                /*neg_a=*/false, a, /*neg_b=*/false, b,
                /*c_mod=*/(short)0, acc[t],
                /*reuse_a=*/false, /*reuse_b=*/false);
        }
    }

    // Store D: VGPR j holds row (row_base + j + 8*hi), col = t*16 + lr.
#pragma unroll
    for (int t = 0; t < 8; ++t) {
        const int col = t * 16 + lr;
        const float bb = bias[col];
#pragma unroll
        for (int j = 0; j < 8; ++j) {
            const int r = row_base + j + hi * 8;
            if (r < nrows) out[(size_t)r * D + col] = acc[t][j] + bb;
        }
    }
}

// ---------------------------------------------------------------------------
// Launch
// ---------------------------------------------------------------------------
extern "C" void kernel_launch(void* const* d_in, const int* in_sizes, int n_in,
                              void* d_out, int out_size, void* d_ws, size_t ws_size,
                              hipStream_t stream) {
    const float* X        = (const float*)d_in[0];
    const int*   edge_row = (const int*)  d_in[1];
    const int*   edge_col = (const int*)  d_in[2];
    const float* edge_val = (const float*)d_in[3];
    const float* Wmat     = (const float*)d_in[4];
    const float* bias     = (const float*)d_in[5];
    float*       out      = (float*)d_out;

    const int N = in_sizes[0] / D;   // number of nodes
    const int E = in_sizes[1];       // number of edges

    // Phase 0: zero agg (= d_out) buffer.
    {
        int n4 = out_size / 4;
        gcn_zero_kernel<<<2048, 256, 0, stream>>>((float4*)out, n4);
    }

    // Phase 1: atomic scatter of messages. One wave per edge.
    {
        long long threads = (long long)E * 32;
        int blocks = (int)((threads + 255) / 256);
        gcn_scatter_kernel<<<blocks, 256, 0, stream>>>(X, edge_row, edge_col,
                                                       edge_val, out, E);
    }

    // Phase 2: in-place WMMA GEMM + bias.
    {
        int blocks = (N + 127) / 128;
        gcn_gemm_wmma_kernel<<<blocks, 256, 0, stream>>>(Wmat, bias, out, N);
    }
}